// BottomUp_86560770883600
// MI455X (gfx1250) — compile-verified
//
#include <hip/hip_runtime.h>
#include <stdint.h>

#ifndef __has_builtin
#define __has_builtin(x) 0
#endif

#define NL 60
#define NC 128

typedef unsigned int u32;
typedef u32 __attribute__((ext_vector_type(4))) u32x4;
typedef int  __attribute__((ext_vector_type(4))) i32x4;
typedef int  __attribute__((ext_vector_type(8))) i32x8;

#if defined(__AMDGCN__) && __has_builtin(__builtin_amdgcn_tensor_load_to_lds) && \
    __has_builtin(__builtin_amdgcn_s_wait_tensorcnt)
#define USE_TDM 1
#else
#define USE_TDM 0
#endif

#if USE_TDM
// DMA one contiguous tile of NL*NC fp32 (30 KB) from global memory into LDS
// using the CDNA5 Tensor Data Mover. Descriptor per CDNA5 ISA.
__device__ __forceinline__ void tdm_load_tile(const float* gsrc, u32 lds_off) {
  const uint64_t ga = (uint64_t)(uintptr_t)gsrc;
  const u32 elems = NL * NC;  // 7680 elements, fits tile_dim0 (16-bit)

  u32x4 g0;
  g0[0] = 1u;                 // count=1, is_restore=0, gather_mode=0
  g0[1] = lds_off;            // lds_addr (bytes)
  g0[2] = (u32)ga;            // global_addr[31:0]
  g0[3] = ((u32)(ga >> 32) & 0x01FFFFFFu) | 0x80000000u;  // addr[56:32] | type=2

  i32x8 g1;
  g1[0] = 0x00020000;               // workgroup_mask=0, data_size=2 (4 bytes)
  g1[1] = (int)(elems << 16);       // tensor_dim0 low16  (bits 63:48)
  g1[2] = 0x00010000;               // tensor_dim0 hi=0, tensor_dim1=1
  g1[3] = (int)(elems << 16);       // tile_dim0 = 7680   (bits 127:112)
  g1[4] = 0;                        // tile_dim1=0 (unused), tile_dim2=0
  g1[5] = (int)elems;               // tensor_dim0_stride low32
  g1[6] = 0;
  g1[7] = 0;

  i32x4 z4 = {0, 0, 0, 0};
#if defined(__clang_major__) && (__clang_major__ >= 23)
  i32x8 z8 = {0, 0, 0, 0, 0, 0, 0, 0};
  __builtin_amdgcn_tensor_load_to_lds(g0, g1, z4, z4, z8, 0);
#else
  __builtin_amdgcn_tensor_load_to_lds(g0, g1, z4, z4, 0);
#endif
}
#endif

// Non-temporal store: outputs are write-once, keep them out of L2's way.
__device__ __forceinline__ void nt_store(float v, float* p) {
  __builtin_nontemporal_store(v, p);
}

__global__ __launch_bounds__(NC, 1)
void adding_doubling_kernel(const float* __restrict__ gA,
                            const float* __restrict__ gR,
                            const float* __restrict__ gT,
                            const float* __restrict__ gS,
                            float* __restrict__ out,
                            size_t plane) {
  // 9 arrays x 30 KB = 270 KB LDS (<= 320 KB/WGP on CDNA5).
  __shared__ float sA[NL * NC];   // absorptance
  __shared__ float sR[NL * NC];   // reflectance
  __shared__ float sT[NL * NC];   // transmittance
  __shared__ float sS[NL * NC];   // source; recycled as absorbed_down
  __shared__ float sRS[NL * NC];  // cumulative surface reflection
  __shared__ float sDS[NL * NC];  // 1/(1 - rs*r)
  __shared__ float sRT[NL * NC];  // cumulative top reflection
  __shared__ float sDT[NL * NC];  // 1/(1 - rt*r)
  __shared__ float sSU[NL * NC];  // multi-scattered upward source

  const int c = threadIdx.x;
  const int b = blockIdx.x;
  const size_t ibase = (size_t)b * (NL * NC);

#if USE_TDM
  if (threadIdx.x < 32u) {  // one wave issues the 4 DMA descriptors (in-order)
    tdm_load_tile(gR + ibase, (u32)(uintptr_t)&sR[0]);  // needed by phase 1
    tdm_load_tile(gT + ibase, (u32)(uintptr_t)&sT[0]);  // needed by phase 1
    tdm_load_tile(gA + ibase, (u32)(uintptr_t)&sA[0]);  // needed by phase 2+
    tdm_load_tile(gS + ibase, (u32)(uintptr_t)&sS[0]);  // needed by phase 2+
    __builtin_amdgcn_s_wait_tensorcnt(2);  // R,T landed; A,S still in flight
  }
#else
#pragma unroll
  for (int l = 0; l < NL; ++l) {
    sA[l * NC + c] = gA[ibase + (size_t)l * NC + c];
    sR[l * NC + c] = gR[ibase + (size_t)l * NC + c];
    sT[l * NC + c] = gT[ibase + (size_t)l * NC + c];
    sS[l * NC + c] = gS[ibase + (size_t)l * NC + c];
  }
#endif
  __syncthreads();

  float* __restrict__ out_fd = out;
  float* __restrict__ out_fu = out + plane;
  float* __restrict__ out_ab = out + 2 * plane;
  const size_t ob = ibase + (size_t)c;

  // ---------- Phase 1: bottom-up cumulative surface reflection ----------
  // (overlaps with the still-in-flight A and S tensor DMAs)
  float rs_next = sR[(NL - 1) * NC + c];
  sRS[(NL - 1) * NC + c] = rs_next;
#pragma unroll
  for (int l = NL - 2; l >= 0; --l) {
    const float rl = sR[l * NC + c];
    const float tl = sT[l * NC + c];
    const float dd = 1.0f / (1.0f - rs_next * rl);
    rs_next = rl + rs_next * tl * tl * dd;
    sRS[l * NC + c] = rs_next;
    sDS[l * NC + c] = dd;
  }

#if USE_TDM
  if (threadIdx.x < 32u) {
    __builtin_amdgcn_s_wait_tensorcnt(0);  // A and S now resident
  }
  __syncthreads();
#endif

  // ---------- Phase 2: top-down rt/dt, s_up, fused downward flux ----------
  float s_lm1 = sS[0 * NC + c];   // s[l-1]
  float r_lm1 = sR[0 * NC + c];   // r[l-1]
  sSU[0 * NC + c] = s_lm1;        // s_up[0] = s[0] (dt[0] = 1)
  sDT[0 * NC + c] = 1.0f;
  sRT[0 * NC + c] = r_lm1;
  float last_rt = r_lm1;
  float flux = 0.0f;
  nt_store(0.0f, &out_fd[ob]);    // flux_down[0]
  sS[0 * NC + c] = 0.0f;          // absorbed_down[0] (s[0] now lives in regs)

#pragma unroll
  for (int l = 1; l < NL; ++l) {
    const float s_l = sS[l * NC + c];  // read before overwrite below
    const float r_l = sR[l * NC + c];
    const float t_l = sT[l * NC + c];
    const float dd = 1.0f / (1.0f - last_rt * r_l);
    sDT[l * NC + c] = dd;
    const float nr = r_l + last_rt * t_l * t_l * dd;
    if (l < NL - 1) sRT[l * NC + c] = nr;
    sSU[l * NC + c] = (s_l + s_lm1 * r_l) * dd;  // s_up[l]
    last_rt = nr;

    // downward-flux scan step, k = l-1 in 0..NL-2
    const float ds_k = sDS[(l - 1) * NC + c];
    flux += (s_lm1 + s_l * r_lm1) * ds_k;        // += s_down[k]
    if (l < NL - 1) {
      const float ds_l = sDS[l * NC + c];
      nt_store(flux, &out_fd[ob + (size_t)l * NC]);  // flux_down[l]
      const float am = sA[l * NC + c] *
                       (1.0f + sRS[(l + 1) * NC + c] * t_l * ds_l);
      sS[l * NC + c] = am * flux;                // absorbed_down[l]
      flux = flux * t_l * ds_l;
    } else {                                     // l == NL-1: fluxN
      nt_store(flux, &out_fd[ob + (size_t)(NL - 1) * NC]);
      sS[(NL - 1) * NC + c] = sA[(NL - 1) * NC + c] * flux;
    }
    s_lm1 = s_l;
    r_lm1 = r_l;
  }

  // ---------- Phase 3: bottom-up upward flux + final absorbed ----------
  float fux = 0.0f;
  nt_store(sS[(NL - 1) * NC + c], &out_ab[ob + (size_t)(NL - 1) * NC]);
#pragma unroll
  for (int l = NL - 1; l >= 2; --l) {
    fux += sSU[l * NC + c];
    nt_store(fux, &out_fu[ob + (size_t)l * NC]);  // flux_up[l]
    const float t1 = sT[(l - 1) * NC + c];
    const float d1 = sDT[(l - 1) * NC + c];
    const float r2 = sRT[(l - 2) * NC + c];
    const float am = sA[(l - 1) * NC + c] * (1.0f + t1 * r2 * d1);
    nt_store(sS[(l - 1) * NC + c] + fux * am,
             &out_ab[ob + (size_t)(l - 1) * NC]);
    fux = fux * t1 * d1;
  }
  const float flux1 = fux + sSU[1 * NC + c];
  nt_store(flux1, &out_fu[ob + (size_t)NC]);                       // flux_up[1]
  nt_store(sS[0 * NC + c] + flux1 * sA[0 * NC + c], &out_ab[ob]);  // absorbed[0]
  nt_store(flux1 * sT[0 * NC + c] + sSU[0 * NC + c], &out_fu[ob]); // flux_up[0]
}

extern "C" void kernel_launch(void* const* d_in, const int* in_sizes, int n_in,
                              void* d_out, int out_size, void* d_ws, size_t ws_size,
                              hipStream_t stream) {
  const float* a = (const float*)d_in[0];
  const float* r = (const float*)d_in[1];
  const float* t = (const float*)d_in[2];
  const float* s = (const float*)d_in[3];
  float* out = (float*)d_out;

  const int B = in_sizes[0] / (NL * NC);       // 2048
  const size_t plane = (size_t)B * NL * NC;    // elements per output tensor

  hipLaunchKernelGGL(adding_doubling_kernel, dim3(B), dim3(NC), 0, stream,
                     a, r, t, s, out, plane);
}